// NonLocalBlock_15101105013204
// MI455X (gfx1250) — compile-verified
//
#include <hip/hip_runtime.h>

typedef float v2f __attribute__((ext_vector_type(2)));
typedef float v8f __attribute__((ext_vector_type(8)));

// D = A(16x4) * B(4x16) + C, full f32 precision on the WMMA pipe
#define WMMA_F32(a, b, c) \
    __builtin_amdgcn_wmma_f32_16x16x4_f32(false, (a), false, (b), (short)0, (c), false, false)

static __device__ __forceinline__ v8f v8zero() {
    v8f z;
    #pragma unroll
    for (int i = 0; i < 8; ++i) z[i] = 0.0f;
    return z;
}

constexpr int kB  = 8;      // batch
constexpr int kC  = 256;    // channels
constexpr int kC2 = 128;    // C/2
constexpr int kH  = 128;
constexpr int kW  = 128;
constexpr int kN  = 16384;  // H*W
constexpr int kG  = 64;     // split-K groups per batch for S

// -------- Kernel 1: projections (theta/phi/g) + S partial accumulation --------
// grid (kG, kB), block 256 (8 waves). Each workgroup owns 256 spatial positions,
// processed as 4 chunks of 64. x is read exactly once by this kernel.
__global__ __launch_bounds__(256) void k_proj(
    const float* __restrict__ x,
    const float* __restrict__ wth, const float* __restrict__ bth,
    const float* __restrict__ wph, const float* __restrict__ bph,
    const float* __restrict__ wgg, const float* __restrict__ bgg,
    float* __restrict__ theta_t,   // [B, N, C2]
    float* __restrict__ Spart)     // [B, G, C2, C2]
{
    __shared__ float Xs[64 * 68];        // x chunk  [k=64][n=64], padded
    __shared__ float Wk[3][128 * 68];    // weight k-chunks [m=128][k=64]
    __shared__ float Ps[128 * 68];       // phi tile [c=128][n=64]
    __shared__ float Gs[128 * 68];       // g   tile [c=128][n=64]
    __shared__ float Ths[64 * 132];      // theta tile transposed [n=64][c=128]
    __shared__ float biasS[3][128];

    const int tid = threadIdx.x;
    const int lane = tid & 31;
    const int wid  = tid >> 5;     // 0..7
    const int lhi  = lane >> 4;    // 0/1 -> K/M half select per ISA layout
    const int l16  = lane & 15;
    const int b  = blockIdx.y;
    const int g  = blockIdx.x;
    const int nbase = g * (kN / kG);   // 256 spatial positions per group

    if (tid < 128) {
        biasS[0][tid] = bth[tid];
        biasS[1][tid] = bph[tid];
        biasS[2][tid] = bgg[tid];
    }

    v8f Sacc[8];
    #pragma unroll
    for (int t = 0; t < 8; ++t) Sacc[t] = v8zero();

    const int m0 = wid * 16;   // this wave's M stripe (c2 rows)

    for (int chunk = 0; chunk < 4; ++chunk) {
        const int n0 = nbase + chunk * 64;

        v8f acc[3][4];
        #pragma unroll
        for (int p = 0; p < 3; ++p)
            #pragma unroll
            for (int t = 0; t < 4; ++t) acc[p][t] = v8zero();

        for (int kb = 0; kb < 4; ++kb) {
            // cooperative load x chunk: 64 rows (k) x 64 cols (n)
            #pragma unroll
            for (int it = 0; it < 4; ++it) {
                int idx = it * 256 + tid;
                int r  = idx >> 4;
                int c4 = (idx & 15) << 2;
                float4 v = *(const float4*)(x + ((size_t)(b * kC + kb * 64 + r)) * kN + n0 + c4);
                *(float4*)(&Xs[r * 68 + c4]) = v;
            }
            // cooperative load weight k-chunks: 128 rows (m) x 64 cols (k) each
            #pragma unroll
            for (int it = 0; it < 8; ++it) {
                int idx = it * 256 + tid;
                int m  = idx >> 4;
                int c4 = (idx & 15) << 2;
                *(float4*)(&Wk[0][m * 68 + c4]) = *(const float4*)(wth + (size_t)m * kC + kb * 64 + c4);
                *(float4*)(&Wk[1][m * 68 + c4]) = *(const float4*)(wph + (size_t)m * kC + kb * 64 + c4);
                *(float4*)(&Wk[2][m * 68 + c4]) = *(const float4*)(wgg + (size_t)m * kC + kb * 64 + c4);
            }
            __syncthreads();

            // GEMM: acc[p] += Wk[p][16-row stripe] @ Xs, K = 64 in steps of 4
            for (int ks = 0; ks < 16; ++ks) {
                const int kk = ks * 4 + lhi * 2;
                const int arow = m0 + l16;
                v2f afr[3];
                #pragma unroll
                for (int p = 0; p < 3; ++p) {
                    afr[p].x = Wk[p][arow * 68 + kk];
                    afr[p].y = Wk[p][arow * 68 + kk + 1];
                }
                #pragma unroll
                for (int tn = 0; tn < 4; ++tn) {
                    const int bn = tn * 16 + l16;
                    v2f bfr;
                    bfr.x = Xs[kk * 68 + bn];
                    bfr.y = Xs[(kk + 1) * 68 + bn];
                    #pragma unroll
                    for (int p = 0; p < 3; ++p)
                        acc[p][tn] = WMMA_F32(afr[p], bfr, acc[p][tn]);
                }
            }
            __syncthreads();
        }

        // Epilogue: stash tiles to LDS (+bias). D layout: VGPR i -> M = m0+i+8*lhi, N = l16.
        #pragma unroll
        for (int tn = 0; tn < 4; ++tn) {
            #pragma unroll
            for (int i = 0; i < 8; ++i) {
                const int cc = m0 + i + lhi * 8;
                const int nn = tn * 16 + l16;
                Ths[nn * 132 + cc] = acc[0][tn][i] + biasS[0][cc];
                Ps[cc * 68 + nn]   = acc[1][tn][i] + biasS[1][cc];
                Gs[cc * 68 + nn]   = acc[2][tn][i] + biasS[2][cc];
            }
        }
        __syncthreads();

        // coalesced write of transposed theta: theta_t[b, n0+n, :]
        #pragma unroll
        for (int it = 0; it < 8; ++it) {
            int idx = it * 256 + tid;
            int nn = idx >> 5;
            int c4 = (idx & 31) << 2;
            float4 v = *(const float4*)(&Ths[nn * 132 + c4]);
            *(float4*)(theta_t + ((size_t)b * kN + n0 + nn) * kC2 + c4) = v;
        }

        // S accumulation: Sacc += phi(16-row stripe) @ g^T, K = 64 spatial
        for (int ks = 0; ks < 16; ++ks) {
            const int kk = ks * 4 + lhi * 2;
            v2f afr;
            afr.x = Ps[(m0 + l16) * 68 + kk];
            afr.y = Ps[(m0 + l16) * 68 + kk + 1];
            #pragma unroll
            for (int tn = 0; tn < 8; ++tn) {
                const int dd = tn * 16 + l16;
                v2f bfr;                     // B[k=n][col=d] = g[d][n]
                bfr.x = Gs[dd * 68 + kk];
                bfr.y = Gs[dd * 68 + kk + 1];
                Sacc[tn] = WMMA_F32(afr, bfr, Sacc[tn]);
            }
        }
        __syncthreads();
    }

    // write deterministic split-K partial of S
    float* P = Spart + ((size_t)(b * kG + g)) * (kC2 * kC2);
    #pragma unroll
    for (int tn = 0; tn < 8; ++tn)
        #pragma unroll
        for (int i = 0; i < 8; ++i) {
            const int cc = m0 + i + lhi * 8;
            const int dd = tn * 16 + l16;
            P[cc * kC2 + dd] = Sacc[tn][i];
        }
}

// -------- Kernel 2: reduce partials + 1/N scale --------
__global__ __launch_bounds__(256) void k_reduce(const float* __restrict__ Spart,
                                                float* __restrict__ S)
{
    const int idx = blockIdx.x * 256 + threadIdx.x;   // 8*16384 total
    const int b = idx >> 14;
    const int e = idx & 16383;
    const float* p = Spart + (size_t)b * kG * 16384 + e;
    float s = 0.0f;
    for (int g = 0; g < kG; ++g) s += p[(size_t)g * 16384];
    S[idx] = s * (1.0f / (float)kN);
}

// -------- Kernel 3: fused y + final 1x1 conv + residual --------
// out[b,o,h,w] = x[b,o,h,w] + b_W[o] + sum_d wW[o,d] * sum_c theta_t[b,d*128+h,c] * S[b,c,w]
// grid (kH, kB), block 256 (8 waves).
__global__ __launch_bounds__(256) void k_out(
    const float* __restrict__ x,
    const float* __restrict__ theta_t,
    const float* __restrict__ S,
    const float* __restrict__ wW, const float* __restrict__ bW,
    float* __restrict__ out)
{
    __shared__ float Ss[128 * 132];   // S_b [c][w]
    __shared__ float Ts[128 * 132];   // T_bh [d][c]; later reused for wW half [o][d]
    __shared__ float Zs[128 * 128];   // Z [d][w]
    __shared__ float Os[128 * 128];   // Out half [o_local][w]

    const int tid = threadIdx.x;
    const int lane = tid & 31;
    const int wid  = tid >> 5;
    const int lhi  = lane >> 4;
    const int l16  = lane & 15;
    const int b = blockIdx.y;
    const int h = blockIdx.x;
    const int m0 = wid * 16;

    const float* Sb = S + (size_t)b * kC2 * kC2;
    #pragma unroll
    for (int it = 0; it < 16; ++it) {
        int idx = it * 256 + tid;
        int r  = idx >> 5;
        int c4 = (idx & 31) << 2;
        *(float4*)(&Ss[r * 132 + c4]) = *(const float4*)(Sb + r * kC2 + c4);
        *(float4*)(&Ts[r * 132 + c4]) =
            *(const float4*)(theta_t + ((size_t)b * kN + r * 128 + h) * kC2 + c4);
    }
    __syncthreads();

    // GEMM1: Z = T_bh @ S_b  (M=d stripe, N=w, K=c=128)
    v8f Z[8];
    #pragma unroll
    for (int t = 0; t < 8; ++t) Z[t] = v8zero();
    for (int ks = 0; ks < 32; ++ks) {
        const int kk = ks * 4 + lhi * 2;
        v2f afr;
        afr.x = Ts[(m0 + l16) * 132 + kk];
        afr.y = Ts[(m0 + l16) * 132 + kk + 1];
        #pragma unroll
        for (int tn = 0; tn < 8; ++tn) {
            const int nn = tn * 16 + l16;
            v2f bfr;
            bfr.x = Ss[kk * 132 + nn];
            bfr.y = Ss[(kk + 1) * 132 + nn];
            Z[tn] = WMMA_F32(afr, bfr, Z[tn]);
        }
    }
    #pragma unroll
    for (int tn = 0; tn < 8; ++tn)
        #pragma unroll
        for (int i = 0; i < 8; ++i)
            Zs[(m0 + i + lhi * 8) * 128 + tn * 16 + l16] = Z[tn][i];
    __syncthreads();

    for (int half = 0; half < 2; ++half) {
        // load wW half (128x128) into Ts
        #pragma unroll
        for (int it = 0; it < 16; ++it) {
            int idx = it * 256 + tid;
            int r  = idx >> 5;
            int c4 = (idx & 31) << 2;
            *(float4*)(&Ts[r * 132 + c4]) =
                *(const float4*)(wW + (size_t)(half * 128 + r) * kC2 + c4);
        }
        __syncthreads();

        // GEMM2: Out = wW_half @ Z  (M=o_local stripe, N=w, K=d=128)
        v8f O[8];
        #pragma unroll
        for (int t = 0; t < 8; ++t) O[t] = v8zero();
        for (int ks = 0; ks < 32; ++ks) {
            const int kk = ks * 4 + lhi * 2;
            v2f afr;
            afr.x = Ts[(m0 + l16) * 132 + kk];
            afr.y = Ts[(m0 + l16) * 132 + kk + 1];
            #pragma unroll
            for (int tn = 0; tn < 8; ++tn) {
                const int nn = tn * 16 + l16;
                v2f bfr;
                bfr.x = Zs[kk * 128 + nn];
                bfr.y = Zs[(kk + 1) * 128 + nn];
                O[tn] = WMMA_F32(afr, bfr, O[tn]);
            }
        }
        #pragma unroll
        for (int tn = 0; tn < 8; ++tn)
            #pragma unroll
            for (int i = 0; i < 8; ++i)
                Os[(m0 + i + lhi * 8) * 128 + tn * 16 + l16] = O[tn][i];
        __syncthreads();

        // coalesced epilogue: residual + bias, full 512B rows
        #pragma unroll
        for (int it = 0; it < 16; ++it) {
            int idx = it * 256 + tid;
            int ol = idx >> 5;
            int c4 = (idx & 31) << 2;
            int o  = half * 128 + ol;
            const float bb = bW[o];
            const size_t base = (((size_t)b * kC + o) * kH + h) * kW + c4;
            float4 xv = *(const float4*)(x + base);
            float4 zv = *(const float4*)(&Os[ol * 128 + c4]);
            float4 r;
            r.x = xv.x + zv.x + bb;
            r.y = xv.y + zv.y + bb;
            r.z = xv.z + zv.z + bb;
            r.w = xv.w + zv.w + bb;
            *(float4*)(out + base) = r;
        }
        __syncthreads();
    }
}

extern "C" void kernel_launch(void* const* d_in, const int* in_sizes, int n_in,
                              void* d_out, int out_size, void* d_ws, size_t ws_size,
                              hipStream_t stream) {
    const float* x   = (const float*)d_in[0];
    const float* wth = (const float*)d_in[1];
    const float* bth = (const float*)d_in[2];
    const float* wph = (const float*)d_in[3];
    const float* bph = (const float*)d_in[4];
    const float* wgg = (const float*)d_in[5];
    const float* bgg = (const float*)d_in[6];
    const float* wW  = (const float*)d_in[7];
    const float* bW  = (const float*)d_in[8];
    float* out = (float*)d_out;

    float* ws = (float*)d_ws;
    float* theta_t = ws;                                   // B*N*C2   = 16,777,216 f
    float* Spart   = ws + (size_t)kB * kN * kC2;           // B*G*C2^2 =  8,388,608 f
    float* S       = Spart + (size_t)kB * kG * kC2 * kC2;  // B*C2^2   =    131,072 f

    k_proj<<<dim3(kG, kB), 256, 0, stream>>>(x, wth, bth, wph, bph, wgg, bgg,
                                             theta_t, Spart);
    k_reduce<<<dim3((kB * kC2 * kC2) / 256), 256, 0, stream>>>(Spart, S);
    k_out<<<dim3(kH, kB), 256, 0, stream>>>(x, theta_t, S, wW, bW, out);
}